// Net_73186242724441
// MI455X (gfx1250) — compile-verified
//
#include <hip/hip_runtime.h>
#include <math.h>

typedef float v2f __attribute__((ext_vector_type(2)));
typedef float v8f __attribute__((ext_vector_type(8)));

#define DEV_INLINE __device__ __forceinline__

// ---------------------------------------------------------------------------
// Degree / normalization kernels
// ---------------------------------------------------------------------------
__global__ void fill_f32(float* __restrict__ p, float v, long long n) {
  long long i = (long long)blockIdx.x * blockDim.x + threadIdx.x;
  if (i < n) p[i] = v;
}

__global__ void deg_count(const int* __restrict__ dst, float* __restrict__ deg, int E) {
  int e = blockIdx.x * blockDim.x + threadIdx.x;
  if (e < E) atomicAdd(&deg[dst[e]], 1.0f);
}

__global__ void deg_to_dis(float* __restrict__ deg, int N) {
  int i = blockIdx.x * blockDim.x + threadIdx.x;
  // self-loop adds 1 to every degree; deg >= 1 so rsqrt(max(deg,1)) == rsqrt(deg)
  if (i < N) deg[i] = rsqrtf(deg[i] + 1.0f);
}

__global__ void edge_norm(const int* __restrict__ src, const int* __restrict__ dst,
                          const float* __restrict__ dis, float* __restrict__ norm, int E) {
  int e = blockIdx.x * blockDim.x + threadIdx.x;
  if (e < E) norm[e] = dis[src[e]] * dis[dst[e]];
}

// ---------------------------------------------------------------------------
// Dense transform: out[N,COUT] = A[N,K] @ W[K,COUT] via V_WMMA_F32_16X16X4_F32.
// 4 waves per block; each wave owns one 16x16 output tile; K fully unrolled.
// Guards are branch-free: loads use clamped in-bounds indices, values are
// zeroed with selects, so no exec-mask branches in the K loop.
// A layout (16x4 f32, ISA 7.12.2): lane L, g=L>>4, m=L&15: a[j]=A[m][k0+2g+j]
// B layout (4x16 f32):            b[j]=W[k0+2g+j][n], n=L&15
// D layout (16x16 f32):           acc[k]=D[8g+k][n]
// ---------------------------------------------------------------------------
template <int K, int COUT>
__global__ __launch_bounds__(128) void gemm_wmma_f32(const float* __restrict__ A,
                                                     const float* __restrict__ W,
                                                     float* __restrict__ out, int MT) {
  const int mt = blockIdx.x * 4 + threadIdx.y;
  if (mt >= MT) return;  // wave-uniform exit; active waves keep EXEC all-ones
  const int lane = threadIdx.x;
  const int g = lane >> 4;
  const int mn = lane & 15;
  const long long row = (long long)mt * 16 + mn;
  const int col = blockIdx.y * 16 + mn;
  const int colc = (col < COUT) ? col : (COUT - 1);  // clamped load index
  const float colw = (col < COUT) ? 1.0f : 0.0f;

  v8f acc = {};
  constexpr int KP = (K + 3) & ~3;
#pragma unroll
  for (int k0 = 0; k0 < KP; k0 += 4) {
    const int ka = k0 + 2 * g;
    const int ka0 = (ka + 0 < K) ? (ka + 0) : (K - 1);
    const int ka1 = (ka + 1 < K) ? (ka + 1) : (K - 1);
    const float w0 = (ka + 0 < K) ? 1.0f : 0.0f;
    const float w1 = (ka + 1 < K) ? 1.0f : 0.0f;
    v2f a, b;
    a.x = A[row * K + ka0] * w0;
    a.y = A[row * K + ka1] * w1;
    b.x = W[ka0 * COUT + colc] * (w0 * colw);
    b.y = W[ka1 * COUT + colc] * (w1 * colw);
    acc = __builtin_amdgcn_wmma_f32_16x16x4_f32(false, a, false, b, (short)0, acc,
                                                false, false);
  }

  if (col < COUT) {
    const long long rbase = (long long)mt * 16 + 8 * g;
#pragma unroll
    for (int k = 0; k < 8; ++k) out[(rbase + k) * COUT + col] = acc[k];
  }
}

// ---------------------------------------------------------------------------
// Aggregation: agg = self-loop init, then scatter-add over edges, then bias+act.
// All vectorized float4 (B128 loads/stores); C is compile-time power of two so
// index math is shifts, no integer division.
// ---------------------------------------------------------------------------
template <int C>
__global__ void self_init(const float* __restrict__ t, const float* __restrict__ dis,
                          float* __restrict__ agg, int N) {
  constexpr int C4 = C / 4;
  long long idx = (long long)blockIdx.x * blockDim.x + threadIdx.x;
  if (idx >= (long long)N * C4) return;
  long long i = idx / C4;
  float d = dis[i];
  float s = d * d;
  float4 tv = reinterpret_cast<const float4*>(t)[idx];
  float4 o = make_float4(tv.x * s, tv.y * s, tv.z * s, tv.w * s);
  reinterpret_cast<float4*>(agg)[idx] = o;  // self-loop message; also zero-init
}

template <int C>
__global__ void edge_aggregate(const float* __restrict__ t, const int* __restrict__ src,
                               const int* __restrict__ dst, const float* __restrict__ norm,
                               float* __restrict__ agg, long long E) {
  constexpr int C4 = C / 4;
  long long idx = (long long)blockIdx.x * blockDim.x + threadIdx.x;
  if (idx >= E * C4) return;
  long long e = idx / C4;            // uniform per wave when C4 >= 32
  int c = (int)(idx % C4) * 4;       // contiguous per lane -> coalesced B128
  int s = src[e];
  int d = dst[e];
  float n = norm[e];
  float4 tv = *reinterpret_cast<const float4*>(&t[(long long)s * C + c]);
  float* ap = &agg[(long long)d * C + c];
  atomicAdd(ap + 0, tv.x * n);  // global_atomic_add_f32
  atomicAdd(ap + 1, tv.y * n);
  atomicAdd(ap + 2, tv.z * n);
  atomicAdd(ap + 3, tv.w * n);
}

template <int C>
__global__ void bias_act(float* __restrict__ buf, const float* __restrict__ b, int N,
                         int relu) {
  constexpr int C4 = C / 4;
  long long idx = (long long)blockIdx.x * blockDim.x + threadIdx.x;
  if (idx >= (long long)N * C4) return;
  int c = (int)(idx % C4) * 4;
  float4 v = reinterpret_cast<const float4*>(buf)[idx];
  v.x += b[c + 0];
  v.y += b[c + 1];
  v.z += b[c + 2];
  v.w += b[c + 3];
  if (relu) {
    v.x = fmaxf(v.x, 0.0f);
    v.y = fmaxf(v.y, 0.0f);
    v.z = fmaxf(v.z, 0.0f);
    v.w = fmaxf(v.w, 0.0f);
  }
  reinterpret_cast<float4*>(buf)[idx] = v;
}

// ---------------------------------------------------------------------------
// Global max pool over batch segments (float atomic max via sign-split trick;
// lowers to native global_atomic_max_i32 / global_atomic_min_u32)
// ---------------------------------------------------------------------------
DEV_INLINE void atomicMaxF(float* addr, float val) {
  if (val >= 0.0f) {
    atomicMax((int*)addr, __float_as_int(val));
  } else {
    atomicMin((unsigned int*)addr, __float_as_uint(val));
  }
}

__global__ void pool_max(const float* __restrict__ h, const int* __restrict__ batch,
                         float* __restrict__ pooled, int N) {
  constexpr int C = 128, C4 = C / 4;
  long long idx = (long long)blockIdx.x * blockDim.x + threadIdx.x;
  if (idx >= (long long)N * C4) return;
  long long i = idx / C4;            // node, uniform per wave
  int c = (int)(idx % C4) * 4;
  int g = batch[i];
  float4 v = reinterpret_cast<const float4*>(h)[idx];
  float* pp = &pooled[(long long)g * C + c];
  atomicMaxF(pp + 0, v.x);
  atomicMaxF(pp + 1, v.y);
  atomicMaxF(pp + 2, v.z);
  atomicMaxF(pp + 3, v.w);
}

// ---------------------------------------------------------------------------
// Head: logits = pooled @ Wl + bl ; log_softmax. 64 graphs, 3 classes.
// ---------------------------------------------------------------------------
__global__ void head(const float* __restrict__ pooled, const float* __restrict__ Wl,
                     const float* __restrict__ bl, float* __restrict__ out, int G) {
  int gidx = blockIdx.x * blockDim.x + threadIdx.x;
  if (gidx >= G) return;
  float l0 = bl[0], l1 = bl[1], l2 = bl[2];
  const float* p = pooled + (long long)gidx * 128;
#pragma unroll 4
  for (int k = 0; k < 128; ++k) {
    float pk = p[k];
    l0 += pk * Wl[k * 3 + 0];
    l1 += pk * Wl[k * 3 + 1];
    l2 += pk * Wl[k * 3 + 2];
  }
  float m = fmaxf(l0, fmaxf(l1, l2));
  float s = __expf(l0 - m) + __expf(l1 - m) + __expf(l2 - m);
  float ls = m + __logf(s);
  out[gidx * 3 + 0] = l0 - ls;
  out[gidx * 3 + 1] = l1 - ls;
  out[gidx * 3 + 2] = l2 - ls;
}

// ---------------------------------------------------------------------------
// Host launcher
// ---------------------------------------------------------------------------
static inline int blocks_for(long long total, int bs) {
  return (int)((total + bs - 1) / bs);
}

template <int K, int COUT>
static void run_layer(const float* h_in, const float* Wm, const float* bm, int relu,
                      const int* src, const int* dst, const float* nrm,
                      const float* dis, float* bufA, float* bufB, int N, int E, int MT,
                      hipStream_t stream) {
  constexpr int NT = (COUT + 15) / 16;
  gemm_wmma_f32<K, COUT>
      <<<dim3((MT + 3) / 4, NT), dim3(32, 4), 0, stream>>>(h_in, Wm, bufA, MT);
  const int BS = 256;
  self_init<COUT>
      <<<blocks_for((long long)N * (COUT / 4), BS), BS, 0, stream>>>(bufA, dis, bufB, N);
  edge_aggregate<COUT>
      <<<blocks_for((long long)E * (COUT / 4), BS), BS, 0, stream>>>(bufA, src, dst, nrm,
                                                                    bufB, E);
  bias_act<COUT>
      <<<blocks_for((long long)N * (COUT / 4), BS), BS, 0, stream>>>(bufB, bm, N, relu);
}

extern "C" void kernel_launch(void* const* d_in, const int* in_sizes, int n_in,
                              void* d_out, int out_size, void* d_ws, size_t ws_size,
                              hipStream_t stream) {
  const float* x = (const float*)d_in[0];  // [N,2]
  const int* ei = (const int*)d_in[1];     // [2,E] flattened
  const int* batch = (const int*)d_in[2];  // [N]
  const float* W1 = (const float*)d_in[3];
  const float* b1 = (const float*)d_in[4];
  const float* W2 = (const float*)d_in[5];
  const float* b2 = (const float*)d_in[6];
  const float* W3 = (const float*)d_in[7];
  const float* b3 = (const float*)d_in[8];
  const float* Wl = (const float*)d_in[9];
  const float* bl = (const float*)d_in[10];
  float* out = (float*)d_out;

  const int N = in_sizes[0] / 2;  // 100000
  const int E = in_sizes[1] / 2;  // 1600000
  const int G = 64;
  const int CMAX = 128;

  const int* src = ei;
  const int* dst = ei + E;

  // workspace layout (floats); all 16B aligned
  float* ws = (float*)d_ws;
  float* dis = ws;                             // N
  float* nrm = dis + N;                        // E
  float* bufA = nrm + E;                       // N*128  (GEMM output t)
  float* bufB = bufA + (long long)N * CMAX;    // N*128  (agg / layer output h)
  float* pooled = bufB + (long long)N * CMAX;  // G*128

  const int BS = 256;

  // --- degree / normalization ------------------------------------------------
  fill_f32<<<blocks_for(N, BS), BS, 0, stream>>>(dis, 0.0f, N);
  deg_count<<<blocks_for(E, BS), BS, 0, stream>>>(dst, dis, E);
  deg_to_dis<<<blocks_for(N, BS), BS, 0, stream>>>(dis, N);
  edge_norm<<<blocks_for(E, BS), BS, 0, stream>>>(src, dst, dis, nrm, E);

  const int MT = N / 16;  // 6250, exact

  // --- three GCN layers (ping-pong: gemm bufB->bufA, aggregate bufA->bufB) ---
  run_layer<2, 8>(x, W1, b1, 1, src, dst, nrm, dis, bufA, bufB, N, E, MT, stream);
  run_layer<8, 32>(bufB, W2, b2, 1, src, dst, nrm, dis, bufA, bufB, N, E, MT, stream);
  run_layer<32, 128>(bufB, W3, b3, 0, src, dst, nrm, dis, bufA, bufB, N, E, MT, stream);

  // --- global max pool + head ------------------------------------------------
  const long long ptot = (long long)G * 128;
  fill_f32<<<blocks_for(ptot, BS), BS, 0, stream>>>(pooled, -INFINITY, ptot);
  pool_max<<<blocks_for((long long)N * 32, BS), BS, 0, stream>>>(bufB, batch, pooled, N);
  head<<<1, 64, 0, stream>>>(pooled, Wl, bl, out, G);
}